// Complex_LSTM_70377334112675
// MI455X (gfx1250) — compile-verified
//
#include <hip/hip_runtime.h>
#include <math.h>

// ---------------------------------------------------------------------------
// Complex LSTM (2 layers, H=512, IN=512, B=64, T=256) for gfx1250 (MI455X)
//
// Strategy:
//   1) deinterleave x -> Xr,Xi planes, rows ordered m = t*B + b
//   2) per layer: one big WMMA f32 complex GEMM (M=16384,K=512,N=2048)
//      computing the input-to-hidden pre-activations for ALL timesteps
//   3) sequential scan: per step a small WMMA GEMM (M=64) accumulates the
//      h2h contribution into the pre-activation rows, then a pointwise
//      kernel applies modulus-gated sigmoid/tanh gates + state update.
// All matrix math uses V_WMMA_F32_16X16X4_F32 (fp32 exact).
// ---------------------------------------------------------------------------

#define TSEQ  256
#define NBAT  64
#define HDIM  512
#define KDIM  512            // IN == H == 512 for both layers
#define NDIM  2048           // 4*H
#define MROWS (TSEQ * NBAT)  // 16384
#define KP    34             // LDS row stride (32 + 2 pad): 8B aligned, conflict-free

typedef __attribute__((ext_vector_type(2))) float v2f;
typedef __attribute__((ext_vector_type(8))) float v8f;

// ---------------------------------------------------------------------------
// x (B,T,IN,2) interleaved -> Xr,Xi planes with row index m = t*B + b
// ---------------------------------------------------------------------------
__global__ __launch_bounds__(256)
void deinterleave_x(const float* __restrict__ x,
                    float* __restrict__ Xr, float* __restrict__ Xi)
{
    const int idx = blockIdx.x * 256 + threadIdx.x;       // over MROWS*KDIM
    const int m = idx / KDIM;                             // m = t*B + b
    const int k = idx - m * KDIM;
    const int t = m / NBAT;
    const int b = m - t * NBAT;
    const float2 v = *(const float2*)(x + ((size_t)(b * TSEQ + t) * KDIM + k) * 2);
    Xr[idx] = v.x;
    Xi[idx] = v.y;
}

// ---------------------------------------------------------------------------
// Complex GEMM with V_WMMA_F32_16X16X4_F32:
//   Cr[m][n] (+)= sum_k Ar[m][k]*Wr[n][k] - Ai[m][k]*Wi[n][k] + br[n]
//   Ci[m][n] (+)= sum_k Ar[m][k]*Wi[n][k] + Ai[m][k]*Wr[n][k] + bi[n]
// W is (N,K) row-major, i.e. already B^T with K contiguous -> B-fragment
// addressing in LDS is identical to the A-fragment addressing.
// Workgroup: 256 threads = 8 waves, 64x64 output tile, K staged 32 at a time.
// Each wave: 2 M-tiles x 1 N-tile, 3 accumulators each (rr, nn=AiBi, ii).
// ---------------------------------------------------------------------------
__global__ __launch_bounds__(256)
void cgemm_wmma(const float* __restrict__ Ar, const float* __restrict__ Ai,
                const float* __restrict__ Wr, const float* __restrict__ Wi,
                const float* __restrict__ br, const float* __restrict__ bi,
                float* __restrict__ Cr, float* __restrict__ Ci,
                int cRowOff, int accumulate)
{
    __shared__ float sAr[64 * KP];
    __shared__ float sAi[64 * KP];
    __shared__ float sBr[64 * KP];
    __shared__ float sBi[64 * KP];

    const int nb    = blockIdx.x * 64;
    const int mb    = blockIdx.y * 64;
    const int tid   = threadIdx.x;
    const int lane  = tid & 31;
    const int wid   = tid >> 5;
    const int wm    = wid >> 2;            // 0..1  (M half of tile)
    const int wn    = wid & 3;             // 0..3  (N quarter of tile)
    const int r16   = lane & 15;
    const int khalf = (lane >> 4) * 2;     // K sub-pair select per ISA layout

    v8f acc_rr[2] = {};   // Ar*Br
    v8f acc_nn[2] = {};   // Ai*Bi (subtracted at the end)
    v8f acc_ii[2] = {};   // Ar*Bi + Ai*Br

    const int lrow = tid >> 2;             // 0..63 : row loaded by this thread
    const int lcol = (tid & 3) * 8;        // 0,8,16,24 : col segment

    for (int kb = 0; kb < KDIM; kb += 32) {
        const size_t aoff = (size_t)(mb + lrow) * KDIM + kb + lcol;
        const size_t boff = (size_t)(nb + lrow) * KDIM + kb + lcol;
        const float4 ar0 = *(const float4*)(Ar + aoff);
        const float4 ar1 = *(const float4*)(Ar + aoff + 4);
        const float4 ai0 = *(const float4*)(Ai + aoff);
        const float4 ai1 = *(const float4*)(Ai + aoff + 4);
        const float4 br0 = *(const float4*)(Wr + boff);
        const float4 br1 = *(const float4*)(Wr + boff + 4);
        const float4 bi0 = *(const float4*)(Wi + boff);
        const float4 bi1 = *(const float4*)(Wi + boff + 4);

        __syncthreads();   // previous iteration's LDS reads done
        *(float4*)&sAr[lrow * KP + lcol]     = ar0;
        *(float4*)&sAr[lrow * KP + lcol + 4] = ar1;
        *(float4*)&sAi[lrow * KP + lcol]     = ai0;
        *(float4*)&sAi[lrow * KP + lcol + 4] = ai1;
        *(float4*)&sBr[lrow * KP + lcol]     = br0;
        *(float4*)&sBr[lrow * KP + lcol + 4] = br1;
        *(float4*)&sBi[lrow * KP + lcol]     = bi0;
        *(float4*)&sBi[lrow * KP + lcol + 4] = bi1;
        __syncthreads();

        #pragma unroll
        for (int kk = 0; kk < 32; kk += 4) {
            const int ko = kk + khalf;
            // A fragments (16x4 f32): lanes 0-15 K={ko,ko+1}, lanes 16-31 K={ko+2,ko+3}
            const v2f a_r0 = *(const v2f*)&sAr[(wm * 32      + r16) * KP + ko];
            const v2f a_r1 = *(const v2f*)&sAr[(wm * 32 + 16 + r16) * KP + ko];
            const v2f a_i0 = *(const v2f*)&sAi[(wm * 32      + r16) * KP + ko];
            const v2f a_i1 = *(const v2f*)&sAi[(wm * 32 + 16 + r16) * KP + ko];
            // B fragments (4x16 f32): same addressing against W rows (n, k)
            const v2f b_r  = *(const v2f*)&sBr[(wn * 16 + r16) * KP + ko];
            const v2f b_i  = *(const v2f*)&sBi[(wn * 16 + r16) * KP + ko];

            acc_rr[0] = __builtin_amdgcn_wmma_f32_16x16x4_f32(false, a_r0, false, b_r, (short)0, acc_rr[0], false, false);
            acc_rr[1] = __builtin_amdgcn_wmma_f32_16x16x4_f32(false, a_r1, false, b_r, (short)0, acc_rr[1], false, false);
            acc_nn[0] = __builtin_amdgcn_wmma_f32_16x16x4_f32(false, a_i0, false, b_i, (short)0, acc_nn[0], false, false);
            acc_nn[1] = __builtin_amdgcn_wmma_f32_16x16x4_f32(false, a_i1, false, b_i, (short)0, acc_nn[1], false, false);
            acc_ii[0] = __builtin_amdgcn_wmma_f32_16x16x4_f32(false, a_r0, false, b_i, (short)0, acc_ii[0], false, false);
            acc_ii[1] = __builtin_amdgcn_wmma_f32_16x16x4_f32(false, a_r1, false, b_i, (short)0, acc_ii[1], false, false);
            acc_ii[0] = __builtin_amdgcn_wmma_f32_16x16x4_f32(false, a_i0, false, b_r, (short)0, acc_ii[0], false, false);
            acc_ii[1] = __builtin_amdgcn_wmma_f32_16x16x4_f32(false, a_i1, false, b_r, (short)0, acc_ii[1], false, false);
        }
    }

    // Epilogue. C/D 16x16 f32 layout: VGPR v -> M = base+v (lanes 0-15) or
    // base+8+v (lanes 16-31); N = base + (lane&15).
    const int   n     = nb + wn * 16 + r16;
    const float biasr = br[n];
    const float biasi = bi[n];
    #pragma unroll
    for (int mt = 0; mt < 2; ++mt) {
        const int m0 = mb + wm * 32 + mt * 16 + (lane >> 4) * 8;
        #pragma unroll
        for (int v = 0; v < 8; ++v) {
            const size_t off = (size_t)(cRowOff + m0 + v) * NDIM + n;
            float vr = acc_rr[mt][v] - acc_nn[mt][v] + biasr;
            float vi = acc_ii[mt][v] + biasi;
            if (accumulate) { vr += Cr[off]; vi += Ci[off]; }
            Cr[off] = vr;
            Ci[off] = vi;
        }
    }
}

// ---------------------------------------------------------------------------
// Zero h, c state
// ---------------------------------------------------------------------------
__global__ __launch_bounds__(256)
void init_state(float* __restrict__ Hr, float* __restrict__ Hi,
                float* __restrict__ Cr, float* __restrict__ Ci)
{
    const int idx = blockIdx.x * 256 + threadIdx.x;   // NBAT*HDIM
    Hr[idx] = 0.0f; Hi[idx] = 0.0f; Cr[idx] = 0.0f; Ci[idx] = 0.0f;
}

// ---------------------------------------------------------------------------
// Pointwise gate math for one timestep.
// pre rows are at m = t*B + b in preR/preI (already i2h + h2h + both biases).
// ---------------------------------------------------------------------------
__device__ __forceinline__ void mod_apply(float pr, float pi, bool sig,
                                          float& outr, float& outi)
{
    const float m = sqrtf(pr * pr + pi * pi);
    const float f = sig ? (1.0f / (1.0f + __expf(-m))) : tanhf(m);
    const float s = f / (m + 1e-8f);
    outr = pr * s;
    outi = pi * s;
}

__global__ __launch_bounds__(256)
void lstm_pointwise(const float* __restrict__ preR, const float* __restrict__ preI,
                    float* __restrict__ Hr, float* __restrict__ Hi,
                    float* __restrict__ Cr, float* __restrict__ Ci,
                    float* __restrict__ Yr, float* __restrict__ Yi,  // layer-0 planes (or null)
                    float* __restrict__ Yint,                        // layer-1 (B,T,H,2) out (or null)
                    int t)
{
    const int idx = blockIdx.x * 256 + threadIdx.x;   // b*HDIM + h
    const int b   = idx >> 9;                          // /512
    const int h   = idx & (HDIM - 1);
    const size_t row = (size_t)(t * NBAT + b) * NDIM;

    float ir, ii, fr, fi, og_r, og_i, gr, gi;
    mod_apply(preR[row + h],            preI[row + h],            true,  ir,   ii);
    mod_apply(preR[row + h + HDIM],     preI[row + h + HDIM],     true,  fr,   fi);
    mod_apply(preR[row + h + 2 * HDIM], preI[row + h + 2 * HDIM], true,  og_r, og_i);
    mod_apply(preR[row + h + 3 * HDIM], preI[row + h + 3 * HDIM], false, gr,   gi);

    // c update is ELEMENTWISE in the reference (not a complex multiply)
    const float cr = Cr[idx] * fr + ir * gr;
    const float ci = Ci[idx] * fi + ii * gi;

    const float m  = sqrtf(cr * cr + ci * ci);
    const float s  = tanhf(m) / (m + 1e-8f);
    const float tr = cr * s;
    const float ti = ci * s;

    // h = o (complex*) mod_tanh(c)
    const float hr = og_r * tr - og_i * ti;
    const float hi = og_i * tr + og_r * ti;

    Cr[idx] = cr; Ci[idx] = ci;
    Hr[idx] = hr; Hi[idx] = hi;

    if (Yr) {  // layer 0: planes feeding layer-1 GEMM (rows t-major)
        const size_t o = (size_t)(t * NBAT + b) * HDIM + h;
        Yr[o] = hr; Yi[o] = hi;
    }
    if (Yint) {  // layer 1: final y (B,T,H,2)
        const size_t o = ((size_t)(b * TSEQ + t) * HDIM + h) * 2;
        Yint[o] = hr; Yint[o + 1] = hi;
    }
}

// ---------------------------------------------------------------------------
// Copy final h,c states into d_out: y | h(2,B,H,2) | c(2,B,H,2)
// ---------------------------------------------------------------------------
__global__ __launch_bounds__(256)
void copy_states(const float* __restrict__ Hr, const float* __restrict__ Hi,
                 const float* __restrict__ Cr, const float* __restrict__ Ci,
                 float* __restrict__ out, int layer)
{
    const int idx = blockIdx.x * 256 + threadIdx.x;   // b*HDIM + h
    const size_t yOff   = (size_t)NBAT * TSEQ * HDIM * 2;      // 16777216
    const size_t lStride = (size_t)NBAT * HDIM * 2;            // 65536
    const size_t hBase = yOff + (size_t)layer * lStride;
    const size_t cBase = yOff + 2 * lStride + (size_t)layer * lStride;
    out[hBase + (size_t)idx * 2]     = Hr[idx];
    out[hBase + (size_t)idx * 2 + 1] = Hi[idx];
    out[cBase + (size_t)idx * 2]     = Cr[idx];
    out[cBase + (size_t)idx * 2 + 1] = Ci[idx];
}

// ---------------------------------------------------------------------------
// Host driver
// ---------------------------------------------------------------------------
extern "C" void kernel_launch(void* const* d_in, const int* in_sizes, int n_in,
                              void* d_out, int out_size, void* d_ws, size_t ws_size,
                              hipStream_t stream)
{
    (void)in_sizes; (void)n_in; (void)out_size; (void)ws_size;

    const float* x = (const float*)d_in[0];
    const float* P[16];
    for (int i = 0; i < 16; ++i) P[i] = (const float*)d_in[1 + i];
    float* out = (float*)d_out;

    // Workspace carve-out (~336 MB)
    char* ws = (char*)d_ws;
    size_t off = 0;
    auto carve = [&](size_t nfloats) {
        float* p = (float*)(ws + off);
        off += nfloats * sizeof(float);
        return p;
    };
    float* Xr   = carve((size_t)MROWS * KDIM);   // also reused as layer-0 output planes
    float* Xi   = carve((size_t)MROWS * KDIM);
    float* preR = carve((size_t)MROWS * NDIM);
    float* preI = carve((size_t)MROWS * NDIM);
    float* Hr   = carve(NBAT * HDIM);
    float* Hi   = carve(NBAT * HDIM);
    float* Cr   = carve(NBAT * HDIM);
    float* Ci   = carve(NBAT * HDIM);

    const dim3 blk(256);

    deinterleave_x<<<(MROWS * KDIM) / 256, blk, 0, stream>>>(x, Xr, Xi);

    for (int l = 0; l < 2; ++l) {
        const float* i2hWr = P[l * 8 + 0];
        const float* i2hWi = P[l * 8 + 1];
        const float* i2hbr = P[l * 8 + 2];
        const float* i2hbi = P[l * 8 + 3];
        const float* h2hWr = P[l * 8 + 4];
        const float* h2hWi = P[l * 8 + 5];
        const float* h2hbr = P[l * 8 + 6];
        const float* h2hbi = P[l * 8 + 7];

        // Big parallel i2h GEMM over all timesteps (overwrites pre buffers)
        cgemm_wmma<<<dim3(NDIM / 64, MROWS / 64), blk, 0, stream>>>(
            Xr, Xi, i2hWr, i2hWi, i2hbr, i2hbi, preR, preI, 0, 0);

        init_state<<<(NBAT * HDIM) / 256, blk, 0, stream>>>(Hr, Hi, Cr, Ci);

        for (int t = 0; t < TSEQ; ++t) {
            // h2h GEMM accumulates into the precomputed pre-activations
            cgemm_wmma<<<dim3(NDIM / 64, 1), blk, 0, stream>>>(
                Hr, Hi, h2hWr, h2hWi, h2hbr, h2hbi, preR, preI, t * NBAT, 1);

            lstm_pointwise<<<(NBAT * HDIM) / 256, blk, 0, stream>>>(
                preR, preI, Hr, Hi, Cr, Ci,
                (l == 0) ? Xr : nullptr, (l == 0) ? Xi : nullptr,
                (l == 1) ? out : nullptr, t);
        }

        copy_states<<<(NBAT * HDIM) / 256, blk, 0, stream>>>(Hr, Hi, Cr, Ci, out, l);
    }
}